// FeatureDictionary_32942399160428
// MI455X (gfx1250) — compile-verified
//
#include <hip/hip_runtime.h>
#include <cstdint>
#include <cstddef>

// Problem constants (from reference): B=4, S=131072, D=64, V=6890
#define NB      4
#define SS      131072
#define NSAMP   (NB * SS)        // 524288 samples
#define DD      64               // feature dim (256 bytes per vertex row)
#define NVERT   6890
#define SAMPLES_PER_TILE  16     // one 256-thread block computes 16 samples/stage
#define TILES_PER_BLOCK   16     // contiguous tiles per block -> deep async pipeline
#define NTILES  (NSAMP / SAMPLES_PER_TILE)          // 32768
#define NBLK    (NTILES / TILES_PER_BLOCK)          // 2048 blocks

// ---------------------------------------------------------------------------
// Main gather + weighted-sum kernel.
// Layout: tid = s*16 + chunk; lane owns float4 chunk `chunk` of sample s's rows.
// LDS: smem[2 buffers][16 samples][3 corner rows][64 floats] = 24 KB.
// Pipeline: async-copy stage k+1's 48 rows into the other buffer while
// computing stage k; each wave issues exactly 3 async loads per stage, so
// `s_wait_asynccnt 3` after issuing the next stage means "previous stage done".
// Each lane consumes exactly the 16-byte chunks it loaded -> no barriers.
// ---------------------------------------------------------------------------
__global__ __launch_bounds__(256)
void fd_feats_kernel(const int* __restrict__ idx,        // [B]
                     const int* __restrict__ smpl_F,     // [NUM_FACES,3]
                     const int* __restrict__ fid,        // [B*S]
                     const float* __restrict__ weights,  // [B*S,3]
                     const float* __restrict__ codebooks,// [256, V, D]
                     float* __restrict__ out_wf)         // [B*S, D]
{
    __shared__ __align__(16) float smem[2 * SAMPLES_PER_TILE * 3 * DD];

    const int tid     = threadIdx.x;
    const int sInTile = tid >> 4;     // 0..15 : which sample of the tile
    const int chunk   = tid & 15;     // 0..15 : which float4 of the 64-f row
    // Low 32 bits of a generic pointer into __shared__ = workgroup-relative
    // LDS byte offset (shared-aperture encoding) -> what VDST expects.
    const uint32_t lds0 = (uint32_t)(uintptr_t)&smem[0];
    const int tile0 = blockIdx.x * TILES_PER_BLOCK;

    auto issue = [&](int tile, int buf) {
        const int sample = tile * SAMPLES_PER_TILE + sInTile;
        const int b      = sample >> 17;                       // S = 2^17
        const float* cb  = codebooks + (size_t)idx[b] * (size_t)(NVERT * DD);
        const int f      = fid[sample];
        const uint32_t lbase =
            lds0 + (uint32_t)((((buf * SAMPLES_PER_TILE + sInTile) * 3) * DD
                               + chunk * 4) * 4);
#pragma unroll
        for (int r = 0; r < 3; ++r) {
            const int v = smpl_F[f * 3 + r];
            const float* g = cb + (size_t)v * DD + chunk * 4;  // 16B aligned
            const uint32_t loff = lbase + (uint32_t)(r * DD * 4);
            asm volatile("global_load_async_to_lds_b128 %0, %1, off"
                         :: "v"(loff), "v"((uint64_t)(uintptr_t)g)
                         : "memory");
        }
    };

    auto compute = [&](int tile, int buf) {
        const int sample = tile * SAMPLES_PER_TILE + sInTile;
        const float* wp = weights + (size_t)sample * 3;
        const float w0 = wp[0], w1 = wp[1], w2 = wp[2];
        const float4* base =
            (const float4*)&smem[((buf * SAMPLES_PER_TILE + sInTile) * 3) * DD];
        const float4 f0 = base[chunk];        // corner 0 row, my chunk
        const float4 f1 = base[16 + chunk];   // corner 1
        const float4 f2 = base[32 + chunk];   // corner 2
        float4 o;
        o.x = fmaf(w2, f2.x, fmaf(w1, f1.x, w0 * f0.x));
        o.y = fmaf(w2, f2.y, fmaf(w1, f1.y, w0 * f0.y));
        o.z = fmaf(w2, f2.z, fmaf(w1, f1.z, w0 * f0.z));
        o.w = fmaf(w2, f2.w, fmaf(w1, f1.w, w0 * f0.w));
        *(float4*)(out_wf + (size_t)sample * DD + chunk * 4) = o;
    };

    // Prologue: stage 0 into buffer 0.
    issue(tile0, 0);
#pragma unroll 1
    for (int it = 0; it < TILES_PER_BLOCK; ++it) {
        if (it + 1 < TILES_PER_BLOCK) {
            issue(tile0 + it + 1, (it + 1) & 1);
            // 3 just-issued loads may remain in flight; previous 3 must land.
            asm volatile("s_wait_asynccnt 3" ::: "memory");
        } else {
            asm volatile("s_wait_asynccnt 0" ::: "memory");
        }
        compute(tile0 + it, it & 1);
    }
}

// ---------------------------------------------------------------------------
// Elementwise tail: normal = (hitpt-coords)/max(||.||,1e-6),
// coords_feats = [w1, w2, sdf]. One thread per sample.
// ---------------------------------------------------------------------------
__global__ __launch_bounds__(256)
void fd_aux_kernel(const float* __restrict__ coords,
                   const float* __restrict__ weights,
                   const float* __restrict__ sdf,
                   const float* __restrict__ hitpt,
                   float* __restrict__ out_cf,
                   float* __restrict__ out_n)
{
    const int i = blockIdx.x * 256 + threadIdx.x;
    if (i >= NSAMP) return;
    const size_t i3 = (size_t)i * 3;
    const float dx = hitpt[i3 + 0] - coords[i3 + 0];
    const float dy = hitpt[i3 + 1] - coords[i3 + 1];
    const float dz = hitpt[i3 + 2] - coords[i3 + 2];
    const float n  = sqrtf(fmaf(dx, dx, fmaf(dy, dy, dz * dz)));
    const float inv = 1.0f / fmaxf(n, 1e-6f);
    out_n[i3 + 0] = dx * inv;
    out_n[i3 + 1] = dy * inv;
    out_n[i3 + 2] = dz * inv;
    out_cf[i3 + 0] = weights[i3 + 1];
    out_cf[i3 + 1] = weights[i3 + 2];
    out_cf[i3 + 2] = sdf[i];
}

extern "C" void kernel_launch(void* const* d_in, const int* in_sizes, int n_in,
                              void* d_out, int out_size, void* d_ws, size_t ws_size,
                              hipStream_t stream) {
    (void)in_sizes; (void)n_in; (void)out_size; (void)d_ws; (void)ws_size;
    const float* coords    = (const float*)d_in[0];
    const int*   idx       = (const int*)  d_in[1];
    const int*   smpl_F    = (const int*)  d_in[2];
    const int*   fid       = (const int*)  d_in[3];
    const float* weights   = (const float*)d_in[4];
    const float* sdf       = (const float*)d_in[5];
    const float* hitpt     = (const float*)d_in[6];
    const float* codebooks = (const float*)d_in[7];

    float* out = (float*)d_out;
    float* out_wf = out;                                   // [B*S,64]
    float* out_cf = out + (size_t)NSAMP * DD;              // [B*S,3]
    float* out_n  = out + (size_t)NSAMP * DD + (size_t)NSAMP * 3;  // [B*S,3]

    fd_feats_kernel<<<NBLK, 256, 0, stream>>>(idx, smpl_F, fid, weights,
                                              codebooks, out_wf);
    fd_aux_kernel<<<(NSAMP + 255) / 256, 256, 0, stream>>>(coords, weights, sdf,
                                                           hitpt, out_cf, out_n);
}